// EvolveGCNO_4896262717837
// MI455X (gfx1250) — compile-verified
//
#include <hip/hip_runtime.h>
#include <hip/hip_bf16.h>
#include <math.h>
#include <stdint.h>

typedef __attribute__((ext_vector_type(2))) float v2f;
typedef __attribute__((ext_vector_type(8))) float v8f;

#define DMODEL 128

__device__ __forceinline__ float sigmoidf_(float x) {
    return 1.0f / (1.0f + __expf(-x));
}

// ---------------------------------------------------------------------------
// Kernel 1: LSTM weight evolution with f32 WMMA.
//   gates = Wg @ w_ih^T + (b_ih + b_hh)   [128 x 512], gate order i,f,g,o
//   c = sigmoid(i) * tanh(g)   (f-gate * c0 == 0, w_hh path == 0 since h0=0)
//   h = sigmoid(o) * tanh(c)   -> Wnew [128,128]
// 8 blocks x 256 threads; block = 16-row tile, each wave = 16-col tile.
// ---------------------------------------------------------------------------
__global__ void evolve_lstm_wmma(const float* __restrict__ Wg,
                                 const float* __restrict__ w_ih,
                                 const float* __restrict__ b_ih,
                                 const float* __restrict__ b_hh,
                                 float* __restrict__ Wnew) {
    __shared__ float sA[16 * DMODEL];  // 8 KB: 16 rows of gcn_weight
    const int tid = threadIdx.x;
    const int r0 = blockIdx.x * 16;

    const float4* src4 = (const float4*)(Wg + (size_t)r0 * DMODEL);
    float4* sA4 = (float4*)sA;
    for (int i = tid; i < 16 * DMODEL / 4; i += blockDim.x) sA4[i] = src4[i];
    __syncthreads();

    const int wave = tid >> 5;
    const int lane = tid & 31;
    const int c0 = wave * 16;          // output-column tile
    const int lm = lane & 15;
    const int kh = (lane >> 4) << 1;   // lanes 0-15: K+0,1 ; lanes 16-31: K+2,3

    v8f ci = {}, cg = {}, co = {};
    // B fragment: B[k][n] = w_ih^T[k][col] = w_ih[col][k]
    const float* bi_row = w_ih + (size_t)(0   + c0 + lm) * DMODEL;  // i rows
    const float* bg_row = w_ih + (size_t)(256 + c0 + lm) * DMODEL;  // g rows
    const float* bo_row = w_ih + (size_t)(384 + c0 + lm) * DMODEL;  // o rows

    for (int k = 0; k < DMODEL; k += 4) {
        v2f a, bi, bg, bo;
        a.x = sA[lm * DMODEL + k + kh];
        a.y = sA[lm * DMODEL + k + kh + 1];
        bi.x = bi_row[k + kh]; bi.y = bi_row[k + kh + 1];
        bg.x = bg_row[k + kh]; bg.y = bg_row[k + kh + 1];
        bo.x = bo_row[k + kh]; bo.y = bo_row[k + kh + 1];
        ci = __builtin_amdgcn_wmma_f32_16x16x4_f32(false, a, false, bi, (short)0, ci, false, false);
        cg = __builtin_amdgcn_wmma_f32_16x16x4_f32(false, a, false, bg, (short)0, cg, false, false);
        co = __builtin_amdgcn_wmma_f32_16x16x4_f32(false, a, false, bo, (short)0, co, false, false);
    }

    // C/D layout: VGPR r -> lanes 0-15: (M=r, N=lane); lanes 16-31: (M=8+r, N=lane-16)
    const int mb = (lane >> 4) << 3;
    const int cc = c0 + lm;
    const float bI = b_ih[cc]       + b_hh[cc];
    const float bG = b_ih[256 + cc] + b_hh[256 + cc];
    const float bO = b_ih[384 + cc] + b_hh[384 + cc];
#pragma unroll
    for (int r = 0; r < 8; ++r) {
        float gi = ci[r] + bI;
        float gg = cg[r] + bG;
        float go = co[r] + bO;
        float cs = sigmoidf_(gi) * tanhf(gg);
        float h  = sigmoidf_(go) * tanhf(cs);
        Wnew[(size_t)(r0 + mb + r) * DMODEL + cc] = h;
    }
}

// ---------------------------------------------------------------------------
// Kernel 2: deg[n] = 1.0 (self-loop weight)
// ---------------------------------------------------------------------------
__global__ void deg_init(float* __restrict__ deg, int n) {
    int i = blockIdx.x * blockDim.x + threadIdx.x;
    if (i < n) deg[i] = 1.0f;
}

// ---------------------------------------------------------------------------
// Kernel 3: deg[col[e]] += ew[e]
// ---------------------------------------------------------------------------
__global__ void deg_scatter(const long long* __restrict__ col,
                            const float* __restrict__ ew,
                            float* __restrict__ deg, int E) {
    int e = blockIdx.x * blockDim.x + threadIdx.x;
    if (e < E) atomicAdd(&deg[col[e]], ew[e]);
}

// ---------------------------------------------------------------------------
// Kernel 4: deg -> dinv = rsqrt(deg) (deg >= 1 always due to self-loop)
// ---------------------------------------------------------------------------
__global__ void deg_to_dinv(float* __restrict__ deg, int n) {
    int i = blockIdx.x * blockDim.x + threadIdx.x;
    if (i < n) {
        float d = deg[i];
        deg[i] = (d > 0.0f) ? rsqrtf(fmaxf(d, 1e-30f)) : 0.0f;
    }
}

// ---------------------------------------------------------------------------
// Kernel 5: xw = X @ Wnew with f32 WMMA.
// Block = 16 rows of X (tile in LDS), 8 waves cover the 8 column tiles.
// ---------------------------------------------------------------------------
__global__ void gemm_xw_wmma(const float* __restrict__ X,
                             const float* __restrict__ W,
                             float* __restrict__ XW) {
    __shared__ float sX[16 * DMODEL];  // 8 KB
    const int tid = threadIdx.x;
    const int m0 = blockIdx.x * 16;

    const float4* src4 = (const float4*)(X + (size_t)m0 * DMODEL);
    float4* sX4 = (float4*)sX;
    for (int i = tid; i < 16 * DMODEL / 4; i += blockDim.x) sX4[i] = src4[i];
    __syncthreads();

    const int wave = tid >> 5;
    const int lane = tid & 31;
    const int n0 = wave * 16;
    const int lm = lane & 15;
    const int kh = (lane >> 4) << 1;

    v8f c = {};
    for (int k = 0; k < DMODEL; k += 4) {
        v2f a, b;
        a.x = sX[lm * DMODEL + k + kh];
        a.y = sX[lm * DMODEL + k + kh + 1];
        b.x = W[(size_t)(k + kh) * DMODEL + n0 + lm];
        b.y = W[(size_t)(k + kh + 1) * DMODEL + n0 + lm];
        c = __builtin_amdgcn_wmma_f32_16x16x4_f32(false, a, false, b, (short)0, c, false, false);
    }

    const int mb = (lane >> 4) << 3;
    const int nn = n0 + lm;
#pragma unroll
    for (int r = 0; r < 8; ++r)
        XW[(size_t)(m0 + mb + r) * DMODEL + nn] = c[r];
}

// ---------------------------------------------------------------------------
// Kernel 6: self-loop term, also initializes d_out:
//   out[n,:] = dinv[n]^2 * xw[n,:]
// ---------------------------------------------------------------------------
__global__ void out_self(const float* __restrict__ xw,
                         const float* __restrict__ dinv,
                         float* __restrict__ out, int n) {
    int i = blockIdx.x * blockDim.x + threadIdx.x;
    if (i < n * DMODEL) {
        int node = i >> 7;
        float di = dinv[node];
        out[i] = di * di * xw[i];
    }
}

// ---------------------------------------------------------------------------
// Kernel 7: edge scatter. One wave32 per edge, float4 per lane (32*4 = 128).
//   out[col,:] += dinv[row]*ew*dinv[col] * xw[row,:]
// ---------------------------------------------------------------------------
__global__ void edge_scatter(const long long* __restrict__ row,
                             const long long* __restrict__ col,
                             const float* __restrict__ ew,
                             const float* __restrict__ dinv,
                             const float* __restrict__ xw,
                             float* __restrict__ out, int E) {
    const int wave = threadIdx.x >> 5;
    const int lane = threadIdx.x & 31;
    const long long e = (long long)blockIdx.x * 8 + wave;
    if (e >= E) return;
    const long long r = row[e];
    const long long c = col[e];
    const float nrm = dinv[r] * ew[e] * dinv[c];
    const float4* src = (const float4*)(xw + r * DMODEL);
    float4 v = src[lane];
    float* dst = out + c * DMODEL + lane * 4;
    atomicAdd(dst + 0, nrm * v.x);
    atomicAdd(dst + 1, nrm * v.y);
    atomicAdd(dst + 2, nrm * v.z);
    atomicAdd(dst + 3, nrm * v.w);
}

// ---------------------------------------------------------------------------
extern "C" void kernel_launch(void* const* d_in, const int* in_sizes, int n_in,
                              void* d_out, int out_size, void* d_ws, size_t ws_size,
                              hipStream_t stream) {
    const float*     X    = (const float*)d_in[0];
    const long long* ei   = (const long long*)d_in[1];   // int64 [2,E]
    const float*     ew   = (const float*)d_in[2];
    const float*     Wg   = (const float*)d_in[3];
    const float*     w_ih = (const float*)d_in[4];
    // d_in[5] = w_hh: h0 == 0, contributes exactly zero -> unused
    const float*     b_ih = (const float*)d_in[6];
    const float*     b_hh = (const float*)d_in[7];
    float*           out  = (float*)d_out;

    const int N = in_sizes[0] / DMODEL;
    const int E = in_sizes[2];
    const long long* rowi = ei;
    const long long* coli = ei + E;

    // Workspace carve-up (256-byte aligned segments)
    float* Wnew = (float*)d_ws;                                        // 128*128
    float* deg  = Wnew + DMODEL * DMODEL;                              // N
    float* xw   = (float*)(((uintptr_t)(deg + N) + 255) & ~(uintptr_t)255);  // N*128

    evolve_lstm_wmma<<<DMODEL / 16, 256, 0, stream>>>(Wg, w_ih, b_ih, b_hh, Wnew);
    deg_init<<<(N + 255) / 256, 256, 0, stream>>>(deg, N);
    deg_scatter<<<(E + 255) / 256, 256, 0, stream>>>(coli, ew, deg, E);
    deg_to_dinv<<<(N + 255) / 256, 256, 0, stream>>>(deg, N);
    gemm_xw_wmma<<<(N + 15) / 16, 256, 0, stream>>>(X, Wnew, xw);
    out_self<<<((size_t)N * DMODEL + 255) / 256, 256, 0, stream>>>(xw, deg, out, N);
    edge_scatter<<<(E + 7) / 8, 256, 0, stream>>>(rowi, coli, ew, deg, xw, out, E);
}